// GraphEncoder_14937896255715
// MI455X (gfx1250) — compile-verified
//
#include <hip/hip_runtime.h>
#include <stdint.h>

// GraphSAGE (3 layers, mean aggregation) for MI455X / gfx1250.
//   per layer: agg = scatter_mean(x[src] -> dst);  out = agg@W_l + b + x@W_r; relu (layers 1,2)
// N = 50000 (divisible by 16), D = 128, E = 640000.

typedef __attribute__((ext_vector_type(2))) float v2f;
typedef __attribute__((ext_vector_type(8))) float v8f;

#define D 128
#define LDS_PITCH 130   // 128 would put all 16 rows on the same LDS bank (stride%64==0)

// ---------------- zero-fill scratch ----------------
__global__ void zero_f32(float* __restrict__ p, long long n) {
    long long i = (long long)blockIdx.x * blockDim.x + threadIdx.x;
    if (i < n) p[i] = 0.0f;
}

// ---------------- in-degree counts (reused by all 3 layers) ----------------
__global__ void sage_count(const long long* __restrict__ dst,
                           float* __restrict__ cnt, int E) {
    int e = blockIdx.x * blockDim.x + threadIdx.x;
    if (e < E) {
        unsafeAtomicAdd(cnt + (int)dst[e], 1.0f);   // GLOBAL_ATOMIC_ADD_F32
    }
}

// ---------------- gather + scatter-add: agg[dst] += h[src] ----------------
// One wave32 per edge; each lane moves 4 consecutive floats (float4 gather,
// 4 hardware f32 atomic adds into L2).
__global__ __launch_bounds__(256) void sage_scatter(
    const float* __restrict__ hin,
    const long long* __restrict__ src,
    const long long* __restrict__ dst,
    float* __restrict__ agg,
    long long nwork)               // E * 32
{
    long long idx = (long long)blockIdx.x * blockDim.x + threadIdx.x;
    if (idx >= nwork) return;
    int e = (int)(idx >> 5);
    int q = ((int)idx & 31) * 4;   // 0,4,...,124
    int s = (int)src[e];
    int d = (int)dst[e];
    const float4 v = *(const float4*)(hin + (size_t)s * D + q);
    float* a = agg + (size_t)d * D + q;
    unsafeAtomicAdd(a + 0, v.x);
    unsafeAtomicAdd(a + 1, v.y);
    unsafeAtomicAdd(a + 2, v.z);
    unsafeAtomicAdd(a + 3, v.w);
}

// ---------------- fused dual-GEMM + bias (+relu) with fp32 WMMA ----------------
// Block: 256 threads = 8 waves. Block b owns rows [16b,16b+16); wave w owns the
// 16x16 output tile at columns [16w,16w+16). K accumulation via
// V_WMMA_F32_16X16X4_F32, 32 steps for agg@W_l + 32 steps for hin@W_r.
// The per-row mean scale 1/max(cnt,1) is folded into the LDS stage of A.
__global__ __launch_bounds__(256) void sage_gemm(
    const float* __restrict__ agg,   // [N,D] raw neighbor sums
    const float* __restrict__ cnt,   // [N]
    const float* __restrict__ hin,   // [N,D]
    const float* __restrict__ Wl,    // [D,D] row-major (k,n)
    const float* __restrict__ Wr,    // [D,D]
    const float* __restrict__ bias,  // [D]
    float* __restrict__ hout,        // [N,D]
    int relu)
{
    __shared__ float s_rs[16];
    __shared__ float s_agg[16 * LDS_PITCH];
    __shared__ float s_hin[16 * LDS_PITCH];

    const int m0 = blockIdx.x * 16;

    if (threadIdx.x < 16) {
        s_rs[threadIdx.x] = 1.0f / fmaxf(cnt[m0 + threadIdx.x], 1.0f);
    }
    __syncthreads();

    // cooperative stage of the 16 A rows (scaled agg + self features)
    for (int i = threadIdx.x; i < 16 * D; i += 256) {
        int r = i >> 7;          // row 0..15
        int k = i & (D - 1);     // 0..127
        s_agg[r * LDS_PITCH + k] = agg[(size_t)(m0 + r) * D + k] * s_rs[r];
        s_hin[r * LDS_PITCH + k] = hin[(size_t)(m0 + r) * D + k];
    }
    __syncthreads();

    const int lane  = threadIdx.x & 31;
    const int wave  = threadIdx.x >> 5;           // 0..7 -> N tile
    const int n0    = wave * 16;
    const int hr    = lane & 15;                  // A row / C column index
    const int col   = n0 + hr;                    // global output column
    const int khalf = (lane < 16) ? 0 : 2;        // A/B K split across lane halves

    float bv = bias[col];
    v8f c = { bv, bv, bv, bv, bv, bv, bv, bv };   // bias preloaded into accumulator

    const float* __restrict__ arow = s_agg + hr * LDS_PITCH;
    #pragma unroll
    for (int k0 = 0; k0 < D; k0 += 4) {
        v2f a, b;
        a[0] = arow[k0 + khalf];
        a[1] = arow[k0 + khalf + 1];
        b[0] = Wl[(size_t)(k0 + khalf) * D + col];
        b[1] = Wl[(size_t)(k0 + khalf + 1) * D + col];
        c = __builtin_amdgcn_wmma_f32_16x16x4_f32(
                false, a, false, b, (short)0, c, false, false);
    }

    const float* __restrict__ hrow = s_hin + hr * LDS_PITCH;
    #pragma unroll
    for (int k0 = 0; k0 < D; k0 += 4) {
        v2f a, b;
        a[0] = hrow[k0 + khalf];
        a[1] = hrow[k0 + khalf + 1];
        b[0] = Wr[(size_t)(k0 + khalf) * D + col];
        b[1] = Wr[(size_t)(k0 + khalf + 1) * D + col];
        c = __builtin_amdgcn_wmma_f32_16x16x4_f32(
                false, a, false, b, (short)0, c, false, false);
    }

    // C/D layout: VGPR i -> M = i (+8 for lanes 16-31), N = lane&15
    const int mbase = m0 + ((lane < 16) ? 0 : 8);
    #pragma unroll
    for (int i = 0; i < 8; ++i) {
        float v = c[i];
        if (relu) v = fmaxf(v, 0.0f);
        hout[(size_t)(mbase + i) * D + col] = v;
    }
}

extern "C" void kernel_launch(void* const* d_in, const int* in_sizes, int n_in,
                              void* d_out, int out_size, void* d_ws, size_t ws_size,
                              hipStream_t stream) {
    (void)n_in; (void)out_size; (void)ws_size;

    const int N = in_sizes[0] / D;        // 50000
    const int E = in_sizes[1] / 2;        // 640000

    const float*     x   = (const float*)d_in[0];
    const long long* ei  = (const long long*)d_in[1];   // int64 edge_index [2,E]
    const long long* src = ei;
    const long long* dst = ei + E;
    const float* Wl1 = (const float*)d_in[2];
    const float* b1  = (const float*)d_in[3];
    const float* Wr1 = (const float*)d_in[4];
    const float* Wl2 = (const float*)d_in[5];
    const float* b2  = (const float*)d_in[6];
    const float* Wr2 = (const float*)d_in[7];
    const float* Wl3 = (const float*)d_in[8];
    const float* b3  = (const float*)d_in[9];
    const float* Wr3 = (const float*)d_in[10];

    float* out = (float*)d_out;

    // workspace layout
    float* agg = (float*)d_ws;                       // N*D
    float* cnt = agg + (size_t)N * D;                // N
    float* h1  = cnt + N;                            // N*D
    float* h2  = h1 + (size_t)N * D;                 // N*D

    const long long nAgg   = (long long)N * D;
    const long long nWork  = (long long)E * 32;
    const int blkZeroAgg   = (int)((nAgg + 255) / 256);
    const int blkScatter   = (int)((nWork + 255) / 256);
    const int blkCount     = (E + 255) / 256;
    const int blkGemm      = N / 16;                 // 3125, exact

    // in-degree counts (edge list is constant across layers)
    zero_f32<<<(N + 255) / 256, 256, 0, stream>>>(cnt, N);
    sage_count<<<blkCount, 256, 0, stream>>>(dst, cnt, E);

    // ---- layer 1: x -> h1 (relu) ----
    zero_f32<<<blkZeroAgg, 256, 0, stream>>>(agg, nAgg);
    sage_scatter<<<blkScatter, 256, 0, stream>>>(x, src, dst, agg, nWork);
    sage_gemm<<<blkGemm, 256, 0, stream>>>(agg, cnt, x, Wl1, Wr1, b1, h1, 1);

    // ---- layer 2: h1 -> h2 (relu) ----
    zero_f32<<<blkZeroAgg, 256, 0, stream>>>(agg, nAgg);
    sage_scatter<<<blkScatter, 256, 0, stream>>>(h1, src, dst, agg, nWork);
    sage_gemm<<<blkGemm, 256, 0, stream>>>(agg, cnt, h1, Wl2, Wr2, b2, h2, 1);

    // ---- layer 3: h2 -> out (no relu) ----
    zero_f32<<<blkZeroAgg, 256, 0, stream>>>(agg, nAgg);
    sage_scatter<<<blkScatter, 256, 0, stream>>>(h2, src, dst, agg, nWork);
    sage_gemm<<<blkGemm, 256, 0, stream>>>(agg, cnt, h2, Wl3, Wr3, b3, out, 0);
}